// Sparse4DUNet_14534169329831
// MI455X (gfx1250) — compile-verified
//
#include <hip/hip_runtime.h>

typedef __attribute__((ext_vector_type(2))) float v2f;
typedef __attribute__((ext_vector_type(8))) float v8f;

#define NEG_SLOPE 0.1f

// ---------------------------------------------------------------- utilities

__global__ __launch_bounds__(256) void zero_f32(float* __restrict__ p, long n) {
  long i = (long)blockIdx.x * blockDim.x + threadIdx.x;
  long st = (long)gridDim.x * blockDim.x;
  for (; i < n; i += st) p[i] = 0.f;
}

__global__ __launch_bounds__(256) void lrelu_k(float* __restrict__ X, long n) {
  long i = (long)blockIdx.x * blockDim.x + threadIdx.x;
  long st = (long)gridDim.x * blockDim.x;
  for (; i < n; i += st) {
    float x = X[i];
    X[i] = (x >= 0.f) ? x : NEG_SLOPE * x;
  }
}

// P[r,0:2] = F[r,0:2]; P[r,2:4] = 0   (pad feats 2ch -> 4ch)
__global__ __launch_bounds__(256) void pad_feats_k(const float* __restrict__ F,
                                                   float* __restrict__ P, long n) {
  long r = (long)blockIdx.x * blockDim.x + threadIdx.x;
  long st = (long)gridDim.x * blockDim.x;
  for (; r < n; r += st) {
    P[r * 4 + 0] = F[r * 2 + 0];
    P[r * 4 + 1] = F[r * 2 + 1];
    P[r * 4 + 2] = 0.f;
    P[r * 4 + 3] = 0.f;
  }
}

// W4[k, kk, c] = (kk < 2) ? W2[k, kk, c] : 0   for 81 offsets, CO=32
__global__ __launch_bounds__(256) void pad_w_k(const float* __restrict__ W2,
                                               float* __restrict__ W4) {
  const long total = 81L * 4 * 32;
  long i = (long)blockIdx.x * blockDim.x + threadIdx.x;
  long st = (long)gridDim.x * blockDim.x;
  for (; i < total; i += st) {
    long k = i / (4 * 32);
    int rem = (int)(i - k * (4 * 32));
    int kk = rem >> 5;
    int c = rem & 31;
    W4[i] = (kk < 2) ? W2[k * 64 + kk * 32 + c] : 0.f;
  }
}

// Y[r, 0:CE] += E[r, 0:CE]  (Y has row stride CY >= CE) : zero-pad skip add
__global__ __launch_bounds__(256) void add_pad_k(float* __restrict__ Y,
                                                 const float* __restrict__ E,
                                                 long n, int CY, int CE) {
  long total = n * CE;
  long i = (long)blockIdx.x * blockDim.x + threadIdx.x;
  long st = (long)gridDim.x * blockDim.x;
  for (; i < total; i += st) {
    long r = i / CE;
    int c = (int)(i - r * CE);
    Y[r * CY + c] += E[i];
  }
}

// ---------------------------------------------------------------- batchnorm

template <int C>
__global__ __launch_bounds__(C) void bn_partial(const float* __restrict__ X, int n,
                                                float* __restrict__ part) {
  int c = threadIdx.x;
  float s = 0.f, s2 = 0.f;
  for (int r = blockIdx.x; r < n; r += gridDim.x) {
    float x = X[(long)r * C + c];
    s += x;
    s2 += x * x;
  }
  part[(long)blockIdx.x * 2 * C + c] = s;
  part[(long)blockIdx.x * 2 * C + C + c] = s2;
}

template <int C>
__global__ __launch_bounds__(C) void bn_finalize(const float* __restrict__ part, int G, int n,
                                                 const float* __restrict__ g,
                                                 const float* __restrict__ b,
                                                 float* __restrict__ ss) {
  int c = threadIdx.x;
  float s = 0.f, s2 = 0.f;
  for (int i = 0; i < G; ++i) {
    s += part[(long)i * 2 * C + c];
    s2 += part[(long)i * 2 * C + C + c];
  }
  float inv_n = 1.f / (float)n;
  float mean = s * inv_n;
  float var = s2 * inv_n - mean * mean;
  float scale = g[c] * rsqrtf(var + 1e-5f);
  ss[c] = scale;
  ss[C + c] = b[c] - mean * scale;
}

__global__ __launch_bounds__(256) void scale_shift_lrelu(float* __restrict__ X,
                                                         const float* __restrict__ ss,
                                                         long total, int C, int doLrelu) {
  long i = (long)blockIdx.x * blockDim.x + threadIdx.x;
  long st = (long)gridDim.x * blockDim.x;
  for (; i < total; i += st) {
    int c = (int)(i % C);
    float y = X[i] * ss[c] + ss[C + c];
    if (doLrelu) y = (y >= 0.f) ? y : NEG_SLOPE * y;
    X[i] = y;
  }
}

// ------------------------------------------------- gather-GEMM-scatter conv
// 4 waves per block; each wave computes a 16-row x CO tile:
//   Y[oi[tile]] += X[ii[tile]] @ Wk
// via V_WMMA_F32_16X16X4_F32 over K in steps of 4, fp32 end-to-end.
// Loads are completely branch-free (tail rows clamp to the last valid row);
// stores take a wave-uniform unguarded fast path for full tiles, so EXEC
// stays all-ones around every WMMA and the common path has no exec juggling.
template <int CI, int CO>
__global__ __launch_bounds__(128) void conv_tile(const float* __restrict__ X,
                                                 const float* __restrict__ Wk,
                                                 const int* __restrict__ ii,
                                                 const int* __restrict__ oi,
                                                 float* __restrict__ Y, int nk) {
  static_assert((CI & 3) == 0, "CI must be a multiple of 4");
  static_assert((CO & 15) == 0, "CO must be a multiple of 16");
  constexpr int NT = CO / 16;
  const int lane = threadIdx.x & 31;
  const int tile = blockIdx.x * 4 + (threadIdx.x >> 5);
  const int ntiles = (nk + 15) >> 4;
  if (tile >= ntiles) return;  // wave-uniform exit

  const int row16 = lane & 15;
  const int half = lane >> 4;
  const int base = tile * 16;

  int r = base + row16;
  if (r > nk - 1) r = nk - 1;  // clamp tail: duplicate last valid row
  const int src = ii[r];
  const float* xrow = X + (long)src * CI;

  v8f acc[NT];
#pragma unroll
  for (int t = 0; t < NT; ++t)
#pragma unroll
    for (int e = 0; e < 8; ++e) acc[t][e] = 0.f;

  for (int k0 = 0; k0 < CI; k0 += 4) {
    const int kk = k0 + 2 * half;  // this lane-half supplies K = kk, kk+1
    v2f a;
    a.x = xrow[kk];
    a.y = xrow[kk + 1];
#pragma unroll
    for (int t = 0; t < NT; ++t) {
      const int col = t * 16 + row16;
      v2f b;
      b.x = Wk[(long)kk * CO + col];
      b.y = Wk[(long)(kk + 1) * CO + col];
      acc[t] = __builtin_amdgcn_wmma_f32_16x16x4_f32(
          false, a, false, b, (short)0, acc[t], false, false);
    }
  }

  // D layout: VGPR v holds row M = v + 8*half, column N = row16
  if (base + 16 <= nk) {
    // full tile: unguarded scatter (common case)
#pragma unroll
    for (int v = 0; v < 8; ++v) {
      const long orow = (long)oi[base + v + 8 * half] * CO;
#pragma unroll
      for (int t = 0; t < NT; ++t) Y[orow + t * 16 + row16] += acc[t][v];
    }
  } else {
    // tail tile: per-row guard
#pragma unroll
    for (int v = 0; v < 8; ++v) {
      const int rr = base + v + 8 * half;
      if (rr < nk) {
        const long orow = (long)oi[rr] * CO;
#pragma unroll
        for (int t = 0; t < NT; ++t) Y[orow + t * 16 + row16] += acc[t][v];
      }
    }
  }
}

// ------------------------------------------------- final projection
// out[r] = sum_c (U1[r,c] + (c<32 ? E0[r,c] : 0)) * wf[c],  C = 64, wave per row
__global__ __launch_bounds__(128) void final_dot(const float* __restrict__ U1,
                                                 const float* __restrict__ E0,
                                                 const float* __restrict__ wf,
                                                 float* __restrict__ out, int n) {
  const int lane = threadIdx.x & 31;
  const int r = blockIdx.x * 4 + (threadIdx.x >> 5);
  if (r >= n) return;
  float a = U1[(long)r * 64 + lane] + E0[(long)r * 32 + lane];  // c = lane (0..31)
  float b = U1[(long)r * 64 + 32 + lane];                       // c = lane+32
  float s = a * wf[lane] + b * wf[32 + lane];
#pragma unroll
  for (int off = 16; off > 0; off >>= 1) s += __shfl_xor(s, off, 32);
  if (lane == 0) out[r] = s;
}

// ---------------------------------------------------------------- host side

static inline int gs_blocks(long n) {
  long b = (n + 255) / 256;
  if (b > 16384) b = 16384;
  if (b < 1) b = 1;
  return (int)b;
}

static inline void zero_buf(float* p, long n, hipStream_t s) {
  zero_f32<<<gs_blocks(n), 256, 0, s>>>(p, n);
}

template <int CI, int CO>
static void run_conv(const float* X, const float* W, int mapBase, int nOff,
                     void* const* d_in, const int* in_sizes, float* Y, hipStream_t s) {
  for (int k = 0; k < nOff; ++k) {
    const int nk = in_sizes[mapBase + 2 * k];
    if (nk <= 0) continue;
    const int* ii = (const int*)d_in[mapBase + 2 * k];
    const int* oi = (const int*)d_in[mapBase + 2 * k + 1];
    const int ntiles = (nk + 15) / 16;
    conv_tile<CI, CO><<<dim3((ntiles + 3) / 4), 128, 0, s>>>(
        X, W + (long)k * CI * CO, ii, oi, Y, nk);
  }
}

template <int C>
static void run_bn(float* X, int n, const float* g, const float* b, bool lrelu,
                   float* part, float* ss, hipStream_t s) {
  bn_partial<C><<<256, C, 0, s>>>(X, n, part);
  bn_finalize<C><<<1, C, 0, s>>>(part, 256, n, g, b, ss);
  const long total = (long)n * C;
  scale_shift_lrelu<<<gs_blocks(total), 256, 0, s>>>(X, ss, total, C, lrelu ? 1 : 0);
}

extern "C" void kernel_launch(void* const* d_in, const int* in_sizes, int n_in,
                              void* d_out, int out_size, void* d_ws, size_t ws_size,
                              hipStream_t stream) {
  (void)n_in; (void)out_size; (void)ws_size;

  const float* feats = (const float*)d_in[0];
  const float* w_e0 = (const float*)d_in[1];
  const float* w_t0 = (const float*)d_in[2];
  const float* w_d1 = (const float*)d_in[3];
  const float* w_t1 = (const float*)d_in[4];
  const float* w_d2 = (const float*)d_in[5];
  const float* w_b1 = (const float*)d_in[6];
  const float* w_b2 = (const float*)d_in[7];
  const float* w_u2 = (const float*)d_in[8];
  const float* w_u1 = (const float*)d_in[9];
  const float* w_fin = (const float*)d_in[10];
  const float* bn0_g = (const float*)d_in[11];
  const float* bn0_b = (const float*)d_in[12];
  const float* bn1_g = (const float*)d_in[13];
  const float* bn1_b = (const float*)d_in[14];
  const float* bn2_g = (const float*)d_in[15];
  const float* bn2_b = (const float*)d_in[16];
  const float* bnb1_g = (const float*)d_in[17];
  const float* bnb1_b = (const float*)d_in[18];
  const float* bnb2_g = (const float*)d_in[19];
  const float* bnb2_b = (const float*)d_in[20];

  // kernel-map flattening: list of (iidx, oidx) pairs, depth-first
  const int ML0 = 21;            // 81 pairs
  const int MD1 = ML0 + 162;     // 81 pairs
  const int ML1 = MD1 + 162;     // 81 pairs
  const int MD2 = ML1 + 162;     // 81 pairs
  const int ML2 = MD2 + 162;     // 81 pairs
  const int MU2 = ML2 + 162;     // 16 pairs
  const int MU1 = MU2 + 32;      // 16 pairs

  const int n0 = in_sizes[0] / 2;
  // center offset (0,0,0,0) of a stride-1 3^4 kernel is index 40 -> hits every site
  const int n1 = in_sizes[ML1 + 2 * 40];
  const int n2 = in_sizes[ML2 + 2 * 40];

  // workspace carve-up
  float* ws = (float*)d_ws;
  size_t off = 0;
  auto carve = [&](size_t nfl) {
    float* p = ws + off;
    off += (nfl + 63) & ~(size_t)63;
    return p;
  };
  float* F4 = carve((size_t)n0 * 4);    // feats padded to 4 channels
  float* W4 = carve((size_t)81 * 4 * 32);  // w_e0 padded to CI=4
  float* A0 = carve((size_t)n0 * 32);   // scratch, level 0
  float* B0 = carve((size_t)n0 * 32);   // e0 (kept for final skip)
  float* C1 = carve((size_t)n1 * 64);   // scratch, level 1
  float* E1 = carve((size_t)n1 * 64);   // e1 (kept for skip)
  float* X2 = carve((size_t)n2 * 128);  // level-2 buffer (96 then 128 ch)
  float* F2 = carve((size_t)n2 * 128);  // level-2 buffer
  float* G1 = carve((size_t)n1 * 96);   // u2 output
  float* U1 = carve((size_t)n0 * 64);   // u1 output
  float* part = carve((size_t)2 * 256 * 128);
  float* ss = carve(256);

  // pad inputs of the first layer so every conv is guard-free (CI % 4 == 0)
  pad_feats_k<<<gs_blocks(n0), 256, 0, stream>>>(feats, F4, n0);
  pad_w_k<<<gs_blocks(81L * 4 * 32), 256, 0, stream>>>(w_e0, W4);

  // encoder level 0
  zero_buf(A0, (long)n0 * 32, stream);
  run_conv<4, 32>(F4, W4, ML0, 81, d_in, in_sizes, A0, stream);
  run_bn<32>(A0, n0, bn0_g, bn0_b, true, part, ss, stream);

  zero_buf(B0, (long)n0 * 32, stream);
  run_conv<32, 32>(A0, w_t0, ML0, 81, d_in, in_sizes, B0, stream);
  lrelu_k<<<gs_blocks((long)n0 * 32), 256, 0, stream>>>(B0, (long)n0 * 32);

  // encoder level 1
  zero_buf(C1, (long)n1 * 64, stream);
  run_conv<32, 64>(B0, w_d1, MD1, 81, d_in, in_sizes, C1, stream);
  run_bn<64>(C1, n1, bn1_g, bn1_b, true, part, ss, stream);

  zero_buf(E1, (long)n1 * 64, stream);
  run_conv<64, 64>(C1, w_t1, ML1, 81, d_in, in_sizes, E1, stream);
  lrelu_k<<<gs_blocks((long)n1 * 64), 256, 0, stream>>>(E1, (long)n1 * 64);

  // encoder level 2
  zero_buf(X2, (long)n2 * 96, stream);
  run_conv<64, 96>(E1, w_d2, MD2, 81, d_in, in_sizes, X2, stream);
  run_bn<96>(X2, n2, bn2_g, bn2_b, true, part, ss, stream);

  // bottleneck
  zero_buf(F2, (long)n2 * 128, stream);
  run_conv<96, 128>(X2, w_b1, ML2, 81, d_in, in_sizes, F2, stream);
  run_bn<128>(F2, n2, bnb1_g, bnb1_b, true, part, ss, stream);

  zero_buf(X2, (long)n2 * 128, stream);
  run_conv<128, 128>(F2, w_b2, ML2, 81, d_in, in_sizes, X2, stream);
  run_bn<128>(X2, n2, bnb2_g, bnb2_b, false, part, ss, stream);

  // decoder: up 2 -> 1
  zero_buf(G1, (long)n1 * 96, stream);
  run_conv<128, 96>(X2, w_u2, MU2, 16, d_in, in_sizes, G1, stream);
  add_pad_k<<<gs_blocks((long)n1 * 64), 256, 0, stream>>>(G1, E1, n1, 96, 64);

  // decoder: up 1 -> 0
  zero_buf(U1, (long)n0 * 64, stream);
  run_conv<96, 64>(G1, w_u1, MU1, 16, d_in, in_sizes, U1, stream);

  // final: (u1 + e0) @ w_fin
  final_dot<<<(n0 + 3) / 4, 128, 0, stream>>>(U1, B0, w_fin, (float*)d_out, n0);
}